// GCAN_18399639896250
// MI455X (gfx1250) — compile-verified
//
#include <hip/hip_runtime.h>
#include <math.h>

// ---------------- problem constants ----------------
#define NN    4000
#define EE    128000
#define ELN   (EE + NN)      // edges incl. self loops
#define LL    15000
#define KDIM  1000           // width of h
#define HEADS 8
#define NHID  64
#define F1    (HEADS * NHID) // 512
#define GHW   64

// ---------------- workspace layout (float offsets) ----------------
static constexpr size_t OFF_H    = 0;          // 4,000,000   h [N,1000]
static constexpr size_t OFF_XL   = 4000000;    // 2,048,000   xl [N,512]
static constexpr size_t OFF_GF   = 6048000;    //   256,000   gf [N,64]
static constexpr size_t OFF_AF1  = 6304000;    //    32,000   per-node feat logits [N,8]
static constexpr size_t OFF_AG1  = 6336000;    //    32,000   per-node gf logits [N,8]
static constexpr size_t OFF_M1   = 6368000;    //    32,000   (uint) seg max
static constexpr size_t OFF_M2   = 6400000;    //    32,000   (uint)
static constexpr size_t OFF_S1   = 6432000;    //    32,000   seg sums
static constexpr size_t OFF_S2   = 6464000;    //    32,000
static constexpr size_t OFF_E1   = 6496000;    // 1,056,000   per-edge exp [E',8]
static constexpr size_t OFF_E2   = 7552000;    // 1,056,000
static constexpr size_t OFF_OUTF = 8608000;    // 2,048,000   aggregated feat [N,512]
static constexpr size_t OFF_OUTG = 10656000;   // 2,048,000   aggregated gf [N,8,64]
static constexpr size_t OFF_XL2  = 12704000;   //     4,000
static constexpr size_t OFF_GF2  = 12708000;   //     4,000
static constexpr size_t OFF_LF2  = 12712000;   //     4,000
static constexpr size_t OFF_LG2  = 12716000;   //     4,000
static constexpr size_t OFF_M1B  = 12720000;   //     4,000   (uint)
static constexpr size_t OFF_M2B  = 12724000;   //     4,000   (uint)
static constexpr size_t OFF_S1B  = 12728000;   //     4,000
static constexpr size_t OFF_S2B  = 12732000;   //     4,000
static constexpr size_t OFF_E1B  = 12736000;   //   132,000
static constexpr size_t OFF_E2B  = 12868000;   //   132,000
static constexpr size_t OFF_OF2  = 13000000;   //     4,000
static constexpr size_t OFF_OG2  = 13004000;   //     4,000
// total: 13,008,000 floats ~= 52 MB

typedef __attribute__((ext_vector_type(2))) float v2f;
typedef __attribute__((ext_vector_type(8))) float v8f;

__device__ __forceinline__ float leaky02(float v) { return v > 0.f ? v : 0.2f * v; }

// order-preserving float->uint encoding for atomicMax-based segment max
__device__ __forceinline__ unsigned fenc(float f) {
  int i = __float_as_int(f);
  return (unsigned)(i ^ ((i >> 31) | 0x80000000));
}
__device__ __forceinline__ float fdec(unsigned u) {
  int i = (u & 0x80000000u) ? (int)(u ^ 0x80000000u) : (int)(~u);
  return __int_as_float(i);
}

// ============ 1) fused Conv1d(4->1,k1) + Conv1d(1->1,k15,s15) ============
// h[n,j] = cj2_b + cj_b*sum_k w2[k] + sum_{c,k} w1[c]*w2[k]*x[n,c,15j+k]
// reads all 960 MB of x exactly once -> bandwidth-bound floor (~41us @ 23.3 TB/s)
__global__ void __launch_bounds__(256) conv_fuse_kernel(
    const float* __restrict__ x, const float* __restrict__ cj_w,
    const float* __restrict__ cj_b, const float* __restrict__ cj2_w,
    const float* __restrict__ cj2_b, float* __restrict__ h) {
  int idx = blockIdx.x * blockDim.x + threadIdx.x;
  if (idx >= NN * KDIM) return;
  int n = idx / KDIM, j = idx - n * KDIM;
  float w2[15];
  float wsum = 0.f;
#pragma unroll
  for (int k = 0; k < 15; ++k) { w2[k] = cj2_w[k]; wsum += w2[k]; }
  const float* xp = x + (size_t)n * 4 * LL + (size_t)j * 15;
  float acc = 0.f;
#pragma unroll
  for (int c = 0; c < 4; ++c) {
    const float* xc = xp + (size_t)c * LL;
    float sc = 0.f;
#pragma unroll
    for (int k = 0; k < 15; ++k) sc += w2[k] * xc[k];
    acc += cj_w[c] * sc;
  }
  h[idx] = acc + cj2_b[0] + cj_b[0] * wsum;
}

// ============ 2) register-blocked f32 WMMA GEMM ============
// C[M,NCOL] = A[M,KD] @ B[KD,NCOL] + bias.  One wave owns a 32x32 macro-tile
// (2x2 grid of 16x16 WMMA tiles, 4 v8f accumulators).  Per k-step: 2 A-frag
// b64 loads + 4 B-frag b32 loads feed 4 v_wmma_f32_16x16x4_f32 -> 1.5 vmem
// per wmma, and NCOL/KD are compile-time so strides fold to immediates.
// Fragment layout per ISA 7.12.2:
//   A (16x4): row = lane&15, VGPR0/1 = K = 2*(lane>>4) + {0,1}
//   B (4x16): col = lane&15, VGPR0/1 = K rows (lane>>4) + {0,2}
//   D (16x16): col = lane&15, VGPR r = row r + 8*(lane>>4)
template <int NCOL, int KD>
__global__ void __launch_bounds__(128) wmma_gemm_bias_kernel(
    const float* __restrict__ A, const float* __restrict__ B,
    const float* __restrict__ bias, float* __restrict__ C) {
  const int lane = threadIdx.x & 31;
  const int wave = threadIdx.x >> 5;
  const int wpb  = blockDim.x >> 5;
  const int tn = (blockIdx.x * wpb + wave) * 32;   // 2 N-tiles per wave
  const int tm = blockIdx.y * 32;                  // 2 M-tiles per wave
  const int mr = lane & 15;
  const int hi = lane >> 4;

  const float* arow0 = A + (size_t)(tm + mr) * KD + 2 * hi;
  const float* arow1 = arow0 + (size_t)16 * KD;
  const float* bcol0 = B + (size_t)hi * NCOL + tn + mr;
  const float* bcol1 = bcol0 + 16;

  v8f acc00 = {}, acc01 = {}, acc10 = {}, acc11 = {};
  for (int k = 0; k < KD; k += 4) {
    v2f a0, a1, b0, b1;
    a0.x = arow0[k];
    a0.y = arow0[k + 1];
    a1.x = arow1[k];
    a1.y = arow1[k + 1];
    b0.x = bcol0[(size_t)k * NCOL];
    b0.y = bcol0[(size_t)(k + 2) * NCOL];
    b1.x = bcol1[(size_t)k * NCOL];
    b1.y = bcol1[(size_t)(k + 2) * NCOL];
    acc00 = __builtin_amdgcn_wmma_f32_16x16x4_f32(false, a0, false, b0,
                                                  (short)0, acc00, false, false);
    acc01 = __builtin_amdgcn_wmma_f32_16x16x4_f32(false, a0, false, b1,
                                                  (short)0, acc01, false, false);
    acc10 = __builtin_amdgcn_wmma_f32_16x16x4_f32(false, a1, false, b0,
                                                  (short)0, acc10, false, false);
    acc11 = __builtin_amdgcn_wmma_f32_16x16x4_f32(false, a1, false, b1,
                                                  (short)0, acc11, false, false);
  }

  const int col0 = tn + mr;
  const int col1 = col0 + 16;
  const float bc0 = bias[col0];
  const float bc1 = bias[col1];
  const int mb0 = tm + hi * 8;
  const int mb1 = mb0 + 16;
#pragma unroll
  for (int r = 0; r < 8; ++r) {
    C[(size_t)(mb0 + r) * NCOL + col0] = acc00[r] + bc0;
    C[(size_t)(mb0 + r) * NCOL + col1] = acc01[r] + bc1;
    C[(size_t)(mb1 + r) * NCOL + col0] = acc10[r] + bc0;
    C[(size_t)(mb1 + r) * NCOL + col1] = acc11[r] + bc1;
  }
}

// ============ 3) per-node attention logit precompute (GAT factorization) ============
__global__ void __launch_bounds__(256) node_logits1_kernel(
    const float* __restrict__ xl, const float* __restrict__ gf,
    const float* __restrict__ att_f, const float* __restrict__ att_g,
    float* __restrict__ af, float* __restrict__ ag) {
  int idx = blockIdx.x * blockDim.x + threadIdx.x; // n*8 + h
  if (idx >= NN * HEADS) return;
  int n = idx >> 3, hh = idx & 7;
  const float* xr = xl + (size_t)n * F1 + hh * NHID;
  const float* gr = gf + (size_t)n * GHW;
  const float* wf = att_f + hh * NHID;
  const float* wg = att_g + hh * GHW;
  float sf = 0.f, sg = 0.f;
#pragma unroll 8
  for (int c = 0; c < NHID; ++c) { sf += xr[c] * wf[c]; sg += gr[c] * wg[c]; }
  af[idx] = sf;
  ag[idx] = sg;
}

__device__ __forceinline__ void edge_sd(const int* __restrict__ src,
                                        const int* __restrict__ dst,
                                        int e, int& s, int& d) {
  if (e < EE) { s = src[e]; d = dst[e]; }
  else        { s = e - EE; d = e - EE; }   // appended self loops
}

// ============ 4) layer-1 segment softmax pass A: atomic max ============
__global__ void __launch_bounds__(256) edge_max1_kernel(
    const int* __restrict__ src, const int* __restrict__ dst,
    const float* __restrict__ af, const float* __restrict__ ag,
    unsigned* __restrict__ m1, unsigned* __restrict__ m2) {
  int idx = blockIdx.x * blockDim.x + threadIdx.x;
  if (idx >= ELN * HEADS) return;
  int e = idx >> 3, hh = idx & 7;
  int s, d; edge_sd(src, dst, e, s, d);
  float a1 = leaky02(af[d * HEADS + hh] + af[s * HEADS + hh]);
  float a2 = leaky02(ag[d * HEADS + hh] + ag[s * HEADS + hh]);
  atomicMax(&m1[d * HEADS + hh], fenc(a1));
  atomicMax(&m2[d * HEADS + hh], fenc(a2));
}

// ============ 5) layer-1 pass B: exp + atomic sum (store per-edge exp) ============
__global__ void __launch_bounds__(256) edge_expsum1_kernel(
    const int* __restrict__ src, const int* __restrict__ dst,
    const float* __restrict__ af, const float* __restrict__ ag,
    const unsigned* __restrict__ m1, const unsigned* __restrict__ m2,
    float* __restrict__ e1b, float* __restrict__ e2b,
    float* __restrict__ s1, float* __restrict__ s2) {
  int idx = blockIdx.x * blockDim.x + threadIdx.x;
  if (idx >= ELN * HEADS) return;
  int e = idx >> 3, hh = idx & 7;
  int s, d; edge_sd(src, dst, e, s, d);
  float a1 = leaky02(af[d * HEADS + hh] + af[s * HEADS + hh]);
  float a2 = leaky02(ag[d * HEADS + hh] + ag[s * HEADS + hh]);
  float x1 = expf(a1 - fdec(m1[d * HEADS + hh]));
  float x2 = expf(a2 - fdec(m2[d * HEADS + hh]));
  e1b[idx] = x1;
  e2b[idx] = x2;
  atomicAdd(&s1[d * HEADS + hh], x1);
  atomicAdd(&s2[d * HEADS + hh], x2);
}

// ============ 6) layer-1 pass C: crossed-weight scatter-add ============
// msg = [ xl[src]*a2_norm , gf[src]*a1_norm ]  -> out[dst]
__global__ void __launch_bounds__(256) edge_aggr1_kernel(
    const int* __restrict__ src, const int* __restrict__ dst,
    const float* __restrict__ xl, const float* __restrict__ gf,
    const float* __restrict__ e1b, const float* __restrict__ e2b,
    const float* __restrict__ s1, const float* __restrict__ s2,
    float* __restrict__ outf, float* __restrict__ outg) {
  __shared__ float sw1[HEADS];
  __shared__ float sw2[HEADS];
  int e = blockIdx.x;
  int s, d; edge_sd(src, dst, e, s, d);
  if (threadIdx.x < HEADS) {
    int hh = threadIdx.x;
    sw1[hh] = e1b[e * HEADS + hh] / (s1[d * HEADS + hh] + 1e-16f);
    sw2[hh] = e2b[e * HEADS + hh] / (s2[d * HEADS + hh] + 1e-16f);
  }
  __syncthreads();
#pragma unroll
  for (int i = 0; i < 4; ++i) {
    int slot = threadIdx.x + i * 256;       // 0..1023 = [h(8)][c(128)]
    int hh = slot >> 7, c = slot & 127;
    if (c < NHID) {
      float v = xl[(size_t)s * F1 + hh * NHID + c] * sw2[hh];
      atomicAdd(&outf[(size_t)(d * HEADS + hh) * NHID + c], v);
    } else {
      int g = c - NHID;
      float v = gf[(size_t)s * GHW + g] * sw1[hh];
      atomicAdd(&outg[(size_t)(d * HEADS + hh) * GHW + g], v);
    }
  }
}

// ============ 7) layer-1 epilogue + layer-2 node precompute ============
// h1=relu(outf+bias1f); xl2 = h1@W2 + b2
// gf1=relu(mean_h outg + bias1g); gf2 = gf1@fc2_w + fc2_b
// lf2 = xl2*att2f, lg2 = gf2*att2g  (per-node logit contributions, heads=1)
__global__ void __launch_bounds__(128) node_layer2_kernel(
    const float* __restrict__ outf, const float* __restrict__ outg,
    const float* __restrict__ bias1f, const float* __restrict__ bias1g,
    const float* __restrict__ W2, const float* __restrict__ b2,
    const float* __restrict__ fc2_w, const float* __restrict__ fc2_b,
    const float* __restrict__ att2f, const float* __restrict__ att2g,
    float* __restrict__ xl2, float* __restrict__ gf2,
    float* __restrict__ lf2, float* __restrict__ lg2) {
  int wave = threadIdx.x >> 5, lane = threadIdx.x & 31;
  int n = blockIdx.x * 4 + wave;           // grid = NN/4 exactly
  float accf = 0.f;
  for (int i = lane; i < F1; i += 32) {
    float h1 = outf[(size_t)n * F1 + i] + bias1f[i];
    h1 = h1 > 0.f ? h1 : 0.f;
    accf += h1 * W2[i];
  }
  float accg = 0.f;
  for (int g = lane; g < GHW; g += 32) {
    float sum = 0.f;
#pragma unroll
    for (int hh = 0; hh < HEADS; ++hh)
      sum += outg[(size_t)(n * HEADS + hh) * GHW + g];
    float gv = sum * (1.f / HEADS) + bias1g[g];
    gv = gv > 0.f ? gv : 0.f;
    accg += gv * fc2_w[g];
  }
#pragma unroll
  for (int off = 16; off > 0; off >>= 1) {
    accf += __shfl_down(accf, off);
    accg += __shfl_down(accg, off);
  }
  if (lane == 0) {
    float xv = accf + b2[0];
    float gv = accg + fc2_b[0];
    xl2[n] = xv;
    gf2[n] = gv;
    lf2[n] = xv * att2f[0];
    lg2[n] = gv * att2g[0];
  }
}

// ============ 8-10) layer-2 edge passes (heads=1, scalar) ============
__global__ void __launch_bounds__(256) edge_max2_kernel(
    const int* __restrict__ src, const int* __restrict__ dst,
    const float* __restrict__ lf2, const float* __restrict__ lg2,
    unsigned* __restrict__ m1, unsigned* __restrict__ m2) {
  int e = blockIdx.x * blockDim.x + threadIdx.x;
  if (e >= ELN) return;
  int s, d; edge_sd(src, dst, e, s, d);
  atomicMax(&m1[d], fenc(leaky02(lf2[d] + lf2[s])));
  atomicMax(&m2[d], fenc(leaky02(lg2[d] + lg2[s])));
}

__global__ void __launch_bounds__(256) edge_expsum2_kernel(
    const int* __restrict__ src, const int* __restrict__ dst,
    const float* __restrict__ lf2, const float* __restrict__ lg2,
    const unsigned* __restrict__ m1, const unsigned* __restrict__ m2,
    float* __restrict__ e1b, float* __restrict__ e2b,
    float* __restrict__ s1, float* __restrict__ s2) {
  int e = blockIdx.x * blockDim.x + threadIdx.x;
  if (e >= ELN) return;
  int s, d; edge_sd(src, dst, e, s, d);
  float x1 = expf(leaky02(lf2[d] + lf2[s]) - fdec(m1[d]));
  float x2 = expf(leaky02(lg2[d] + lg2[s]) - fdec(m2[d]));
  e1b[e] = x1;
  e2b[e] = x2;
  atomicAdd(&s1[d], x1);
  atomicAdd(&s2[d], x2);
}

__global__ void __launch_bounds__(256) edge_aggr2_kernel(
    const int* __restrict__ src, const int* __restrict__ dst,
    const float* __restrict__ xl2, const float* __restrict__ gf2,
    const float* __restrict__ e1b, const float* __restrict__ e2b,
    const float* __restrict__ s1, const float* __restrict__ s2,
    float* __restrict__ of2, float* __restrict__ og2) {
  int e = blockIdx.x * blockDim.x + threadIdx.x;
  if (e >= ELN) return;
  int s, d; edge_sd(src, dst, e, s, d);
  float w1 = e1b[e] / (s1[d] + 1e-16f);
  float w2 = e2b[e] / (s2[d] + 1e-16f);
  atomicAdd(&of2[d], xl2[s] * w2);   // crossed weighting
  atomicAdd(&og2[d], gf2[s] * w1);
}

// ============ 11) final sigmoid ============
__global__ void __launch_bounds__(256) final_kernel(
    const float* __restrict__ of2, const float* __restrict__ og2,
    const float* __restrict__ bias2f, const float* __restrict__ bias2g,
    float* __restrict__ out) {
  int n = blockIdx.x * blockDim.x + threadIdx.x;
  if (n >= NN) return;
  float z = (of2[n] + bias2f[0]) + (og2[n] + bias2g[0]);
  out[n] = 1.f / (1.f + expf(-z));
}

// =====================================================================
extern "C" void kernel_launch(void* const* d_in, const int* in_sizes, int n_in,
                              void* d_out, int out_size, void* d_ws, size_t ws_size,
                              hipStream_t stream) {
  const float* x        = (const float*)d_in[0];
  const float* geneflow = (const float*)d_in[1];
  const int*   eidx     = (const int*)d_in[2];
  const float* cj_w   = (const float*)d_in[3];
  const float* cj_b   = (const float*)d_in[4];
  const float* cj2_w  = (const float*)d_in[5];
  const float* cj2_b  = (const float*)d_in[6];
  const float* fc1_w  = (const float*)d_in[7];
  const float* fc1_b  = (const float*)d_in[8];
  const float* fc2_w  = (const float*)d_in[9];
  const float* fc2_b  = (const float*)d_in[10];
  const float* W1     = (const float*)d_in[11];
  const float* b1     = (const float*)d_in[12];
  const float* att1f  = (const float*)d_in[13];
  const float* att1g  = (const float*)d_in[14];
  const float* bias1f = (const float*)d_in[15];
  const float* bias1g = (const float*)d_in[16];
  const float* W2     = (const float*)d_in[17];
  const float* b2     = (const float*)d_in[18];
  const float* att2f  = (const float*)d_in[19];
  const float* att2g  = (const float*)d_in[20];
  const float* bias2f = (const float*)d_in[21];
  const float* bias2g = (const float*)d_in[22];

  const int* esrc = eidx;        // edge_index[0]
  const int* edst = eidx + EE;   // edge_index[1]

  float* ws = (float*)d_ws;
  float* h    = ws + OFF_H;
  float* xl   = ws + OFF_XL;
  float* gf   = ws + OFF_GF;
  float* af1  = ws + OFF_AF1;
  float* ag1  = ws + OFF_AG1;
  unsigned* m1 = (unsigned*)(ws + OFF_M1);
  unsigned* m2 = (unsigned*)(ws + OFF_M2);
  float* s1   = ws + OFF_S1;
  float* s2   = ws + OFF_S2;
  float* e1b  = ws + OFF_E1;
  float* e2b  = ws + OFF_E2;
  float* outf = ws + OFF_OUTF;
  float* outg = ws + OFF_OUTG;
  float* xl2  = ws + OFF_XL2;
  float* gf2  = ws + OFF_GF2;
  float* lf2  = ws + OFF_LF2;
  float* lg2  = ws + OFF_LG2;
  unsigned* m1b = (unsigned*)(ws + OFF_M1B);
  unsigned* m2b = (unsigned*)(ws + OFF_M2B);
  float* s1b  = ws + OFF_S1B;
  float* s2b  = ws + OFF_S2B;
  float* e1c  = ws + OFF_E1B;
  float* e2c  = ws + OFF_E2B;
  float* of2  = ws + OFF_OF2;
  float* og2  = ws + OFF_OG2;
  float* out  = (float*)d_out;

  // zero accumulators (encoded-uint 0 is below fenc(-inf), so valid max init)
  hipMemsetAsync(ws + OFF_M1,   0, 4 * 32000 * sizeof(float), stream);       // m1,m2,s1,s2
  hipMemsetAsync(ws + OFF_OUTF, 0, 4096000 * sizeof(float), stream);         // outf,outg
  hipMemsetAsync(ws + OFF_M1B,  0, 16000 * sizeof(float), stream);           // m1b..s2b
  hipMemsetAsync(ws + OFF_OF2,  0, 8000 * sizeof(float), stream);            // of2,og2

  // 1) fused conv: 960 MB read (bandwidth floor of the whole pipeline)
  conv_fuse_kernel<<<(NN * KDIM + 255) / 256, 256, 0, stream>>>(
      x, cj_w, cj_b, cj2_w, cj2_b, h);

  // 2) gf = geneflow @ fc1_w + fc1_b   (WMMA f32; N=64 -> 2 waves of 32 cols)
  wmma_gemm_bias_kernel<GHW, 64><<<dim3(1, NN / 32), 64, 0, stream>>>(
      geneflow, fc1_w, fc1_b, gf);

  // 3) xl = h @ W1 + b1   (4000x1000x512 WMMA f32, 32x32 macro-tile per wave)
  wmma_gemm_bias_kernel<F1, KDIM><<<dim3(F1 / 128, NN / 32), 128, 0, stream>>>(
      h, W1, b1, xl);

  // 4) per-node attention logits
  node_logits1_kernel<<<(NN * HEADS + 255) / 256, 256, 0, stream>>>(
      xl, gf, att1f, att1g, af1, ag1);

  // 5-7) layer-1 segment softmax + scatter aggregation
  edge_max1_kernel<<<(ELN * HEADS + 255) / 256, 256, 0, stream>>>(
      esrc, edst, af1, ag1, m1, m2);
  edge_expsum1_kernel<<<(ELN * HEADS + 255) / 256, 256, 0, stream>>>(
      esrc, edst, af1, ag1, m1, m2, e1b, e2b, s1, s2);
  edge_aggr1_kernel<<<ELN, 256, 0, stream>>>(
      esrc, edst, xl, gf, e1b, e2b, s1, s2, outf, outg);

  // 8) layer-1 epilogue + layer-2 per-node scalars (wave per node)
  node_layer2_kernel<<<NN / 4, 128, 0, stream>>>(
      outf, outg, bias1f, bias1g, W2, b2, fc2_w, fc2_b, att2f, att2g,
      xl2, gf2, lf2, lg2);

  // 9-11) layer-2 edge passes
  edge_max2_kernel<<<(ELN + 255) / 256, 256, 0, stream>>>(
      esrc, edst, lf2, lg2, m1b, m2b);
  edge_expsum2_kernel<<<(ELN + 255) / 256, 256, 0, stream>>>(
      esrc, edst, lf2, lg2, m1b, m2b, e1c, e2c, s1b, s2b);
  edge_aggr2_kernel<<<(ELN + 255) / 256, 256, 0, stream>>>(
      esrc, edst, xl2, gf2, e1c, e2c, s1b, s2b, of2, og2);

  // 12) output
  final_kernel<<<(NN + 255) / 256, 256, 0, stream>>>(
      of2, og2, bias2f, bias2g, out);
}